// GCN_VI_58248346468476
// MI455X (gfx1250) — compile-verified
//
#include <hip/hip_runtime.h>
#include <math.h>

#define N_NODES 40000
#define C_IN    128
#define H_DIM   128

typedef float v2f __attribute__((ext_vector_type(2)));
typedef float v8f __attribute__((ext_vector_type(8)));

// ---------- degree / normalization ----------
__global__ void deg_init_kernel(float* __restrict__ deg, int n) {
    int i = blockIdx.x * blockDim.x + threadIdx.x;
    if (i < n) deg[i] = 1.0f;  // self-loop
}

__global__ void deg_count_kernel(const int* __restrict__ dst, float* __restrict__ deg, int E) {
    int e = blockIdx.x * blockDim.x + threadIdx.x;
    if (e < E) atomicAdd(&deg[dst[e]], 1.0f);
}

__global__ void dinv_kernel(float* __restrict__ deg, int n) {
    int i = blockIdx.x * blockDim.x + threadIdx.x;
    if (i < n) {
        float d = deg[i];
        deg[i] = (d > 0.0f) ? rsqrtf(d) : 0.0f;  // in-place: deg -> dinv
    }
}

// ---------- layer 1 GEMM: g = (x @ W1^T) * dinv[row]; also acc1 = g (self-loop init) ----------
// One wave per 16x16 output tile; 8 waves/block cover all 8 N-tiles of one M-tile.
__global__ __launch_bounds__(256) void gemm1_wmma_kernel(
    const float* __restrict__ x, const float* __restrict__ W1,
    const float* __restrict__ dinv,
    float* __restrict__ g, float* __restrict__ acc1) {
    const int lane = threadIdx.x & 31;
    const int nt   = threadIdx.x >> 5;   // 0..7  -> N tile
    const int mt   = blockIdx.x;         // 0..2499 -> M tile
    const int half = lane >> 4;          // K sub-select
    const int r    = lane & 15;          // M (for A) / N (for B) index

    const float* arow = x  + (size_t)(mt * 16 + r) * C_IN + 2 * half;
    const float* brow = W1 + (size_t)(nt * 16 + r) * C_IN + 2 * half;

    v8f acc = {};
#pragma unroll
    for (int k0 = 0; k0 < C_IN; k0 += 4) {
        v2f a = *(const v2f*)(arow + k0);
        v2f b = *(const v2f*)(brow + k0);
        acc = __builtin_amdgcn_wmma_f32_16x16x4_f32(
            /*neg_a=*/false, a, /*neg_b=*/false, b,
            /*c_mod=*/(short)0, acc, /*reuse_a=*/false, /*reuse_b=*/false);
    }

    const int ncol = nt * 16 + r;
#pragma unroll
    for (int v = 0; v < 8; ++v) {
        int row = mt * 16 + v + 8 * half;
        float val = acc[v] * dinv[row];
        size_t idx = (size_t)row * H_DIM + ncol;
        g[idx]    = val;
        acc1[idx] = val;  // self-loop contribution pre-seeded
    }
}

// ---------- layer 1 scatter: acc1[dst] += g[src]  (one wave per edge, 4 floats/lane) ----------
__global__ __launch_bounds__(256) void scatter1_kernel(
    const int* __restrict__ src, const int* __restrict__ dst,
    const float* __restrict__ g, float* __restrict__ acc1, int E) {
    long long tid = (long long)blockIdx.x * blockDim.x + threadIdx.x;
    int e    = (int)(tid >> 5);
    int lane = (int)(tid & 31);
    if (e >= E) return;
    int s = src[e];
    int d = dst[e];
    const float4 v = *(const float4*)(g + (size_t)s * H_DIM + lane * 4);
    float* p = acc1 + (size_t)d * H_DIM + lane * 4;
    atomicAdd(p + 0, v.x);
    atomicAdd(p + 1, v.y);
    atomicAdd(p + 2, v.z);
    atomicAdd(p + 3, v.w);
}

// ---------- layer 1 epilogue: h1 = relu(dinv[n]*acc1 + b1)  (written over g) ----------
__global__ void finish1_kernel(const float* __restrict__ acc1, const float* __restrict__ dinv,
                               const float* __restrict__ b1, float* __restrict__ h1, int n) {
    int i = blockIdx.x * blockDim.x + threadIdx.x;
    if (i < n) {
        int node = i >> 7;        // /128
        int j    = i & 127;
        float v = dinv[node] * acc1[i] + b1[j];
        h1[i] = fmaxf(v, 0.0f);
    }
}

// ---------- layer 2 dot: g2[n] = dinv[n] * dot(h1[n], W2); acc2 seeded with self-loop ----------
__global__ __launch_bounds__(256) void layer2_dot_kernel(
    const float* __restrict__ h1, const float* __restrict__ W2,
    const float* __restrict__ dinv, float* __restrict__ g2, float* __restrict__ acc2, int N) {
    long long tid = (long long)blockIdx.x * blockDim.x + threadIdx.x;
    int n    = (int)(tid >> 5);
    int lane = (int)(tid & 31);
    if (n >= N) return;
    float4 hv = *(const float4*)(h1 + (size_t)n * H_DIM + lane * 4);
    float4 wv = *(const float4*)(W2 + lane * 4);
    float p = hv.x * wv.x + hv.y * wv.y + hv.z * wv.z + hv.w * wv.w;
#pragma unroll
    for (int off = 16; off > 0; off >>= 1) p += __shfl_xor(p, off, 32);
    if (lane == 0) {
        float z = p * dinv[n];
        g2[n]   = z;
        acc2[n] = z;  // self-loop
    }
}

// ---------- layer 2 scatter: acc2[dst] += g2[src] ----------
__global__ void scatter2_kernel(const int* __restrict__ src, const int* __restrict__ dst,
                                const float* __restrict__ g2, float* __restrict__ acc2, int E) {
    int e = blockIdx.x * blockDim.x + threadIdx.x;
    if (e < E) atomicAdd(&acc2[dst[e]], g2[src[e]]);
}

// ---------- final: out = sigmoid(dinv[n]*acc2[n] + b2) ----------
__global__ void final_kernel(const float* __restrict__ acc2, const float* __restrict__ dinv,
                             const float* __restrict__ b2, float* __restrict__ out, int N) {
    int n = blockIdx.x * blockDim.x + threadIdx.x;
    if (n < N) {
        float z = dinv[n] * acc2[n] + b2[0];
        out[n] = 1.0f / (1.0f + expf(-z));
    }
}

extern "C" void kernel_launch(void* const* d_in, const int* in_sizes, int n_in,
                              void* d_out, int out_size, void* d_ws, size_t ws_size,
                              hipStream_t stream) {
    const float* x  = (const float*)d_in[0];
    const int*   ei = (const int*)d_in[1];   // [2, E] row-major: src then dst
    const float* W1 = (const float*)d_in[2]; // [128,128]
    const float* b1 = (const float*)d_in[3]; // [128]
    const float* W2 = (const float*)d_in[4]; // [1,128]
    const float* b2 = (const float*)d_in[5]; // [1]
    float* out = (float*)d_out;

    const int N = N_NODES;
    const int E = in_sizes[1] / 2;
    const int* src = ei;
    const int* dst = ei + E;

    // workspace layout (floats)
    float* ws   = (float*)d_ws;
    float* dinv = ws;                            // N   (deg -> dinv in place)
    float* g    = dinv + N;                      // N*128 (also reused as h1)
    float* acc1 = g    + (size_t)N * H_DIM;      // N*128
    float* g2   = acc1 + (size_t)N * H_DIM;      // N
    float* acc2 = g2   + N;                      // N

    const int T = 256;
    deg_init_kernel <<<(N + T - 1) / T, T, 0, stream>>>(dinv, N);
    deg_count_kernel<<<(E + T - 1) / T, T, 0, stream>>>(dst, dinv, E);
    dinv_kernel     <<<(N + T - 1) / T, T, 0, stream>>>(dinv, N);

    gemm1_wmma_kernel<<<N / 16, 256, 0, stream>>>(x, W1, dinv, g, acc1);

    long long s1_threads = (long long)E * 32;
    scatter1_kernel<<<(unsigned)((s1_threads + T - 1) / T), T, 0, stream>>>(src, dst, g, acc1, E);

    finish1_kernel<<<(N * H_DIM + T - 1) / T, T, 0, stream>>>(acc1, dinv, b1, g, N * H_DIM);

    long long l2_threads = (long long)N * 32;
    layer2_dot_kernel<<<(unsigned)((l2_threads + T - 1) / T), T, 0, stream>>>(g, W2, dinv, g2, acc2, N);

    scatter2_kernel<<<(E + T - 1) / T, T, 0, stream>>>(src, dst, g2, acc2, E);

    final_kernel<<<(N + T - 1) / T, T, 0, stream>>>(acc2, dinv, b2, out, N);
}